// CustomLoss_7060926235048
// MI455X (gfx1250) — compile-verified
//
#include <hip/hip_runtime.h>
#include <hip/hip_bf16.h>
#include <stdint.h>

typedef float v2f __attribute__((ext_vector_type(2)));
typedef float v8f __attribute__((ext_vector_type(8)));

__device__ __forceinline__ float sigmoidf(float x) {
    return 1.0f / (1.0f + __expf(-x));
}

// Accumulator slots per batch in ws:
// 0: sum fg   1: sum gt_fg   2: sum fg*gt_fg
// 3: sum bg   4: sum gt_bg   5: sum bg*gt_bg
// 6: sum surf*gt_area        7: sum (surf+gt_surf)*gt_area
__launch_bounds__(256)
__global__ void surf_dice_main(const float* __restrict__ preds,
                               const float* __restrict__ targets,
                               const float* __restrict__ area,
                               float* __restrict__ ws) {
    constexpr int D = 128, H = 256, W = 256;
    constexpr int DS = 4;   // output d-slices per block
    constexpr int HC = 32;  // output h-rows per block

    // Row staging, double buffered: [5 slices raw preds][5 slices targets], W=256 each.
    __shared__ __align__(16) float buf[2][2560];  // 20 KB
    __shared__ float areaS[256];
    __shared__ float Pws[8][8];                   // per-wave partial sums [wave][acc]

    const int t  = threadIdx.x;
    const int bi = blockIdx.x;                 // 512 blocks total
    const int b  = bi >> 8;                    // batch
    const int r  = bi & 255;
    const int d0 = (r >> 3) * DS;              // first output d of slab
    const int h0 = (r & 7) * HC;               // first input row of chunk
    const int hLast = min(h0 + HC, H - 1);     // last input row (inclusive)
    const uint32_t bOff = (uint32_t)b * (uint32_t)(D * H * W);

    areaS[t] = area[t];

    // Issue async global->LDS b128 copies for input row hn.
    // Row payload = 640 float4 (5 slices preds + 5 slices targets): k=0,1 full, k=2 half.
    auto issueRow = [&](int hn) {
        const uint32_t bufb = (uint32_t)(uintptr_t)(&buf[hn & 1][0]);
#pragma unroll
        for (int k = 0; k < 3; ++k) {
            if (k < 2 || t < 128) {
                const int f   = (k << 8) + t;            // float4 index 0..639
                const int isT = (f >= 320);
                const int rr  = isT ? (f - 320) : f;
                const int j   = rr >> 6;                 // slice within slab
                const int w4  = (rr & 63) << 2;          // starting w
                int dd = d0 + j; if (dd > D - 1) dd = D - 1;
                const uint32_t goff  = (bOff + ((uint32_t)dd << 16) +
                                        ((uint32_t)hn << 8) + (uint32_t)w4) << 2;
                const uint32_t laddr = bufb + ((uint32_t)f << 4);
                const float*   base  = isT ? targets : preds;
                asm volatile("global_load_async_to_lds_b128 %0, %1, %2"
                             :: "v"(laddr), "v"(goff), "s"(base) : "memory");
            }
        }
    };

    float rfp[DS], rbp[DS];
    int   ap[DS];
    float acc[8] = {0.f, 0.f, 0.f, 0.f, 0.f, 0.f, 0.f, 0.f};

    issueRow(h0);

    for (int h = h0; h <= hLast; ++h) {
        // Wait for row h (issued last iteration), then one barrier per row.
        asm volatile("s_wait_asynccnt 0x0" ::: "memory");
        __syncthreads();

        // Prefetch next row into the other buffer; overlaps with compute below.
        if (h < hLast) issueRow(h + 1);

        const int w = t;
        if (w < W - 1) {
            const float* P = &buf[h & 1][0];
            const float* T = P + 1280;
            float p_[5], q_[5], c_[5];
#pragma unroll
            for (int j = 0; j < 5; ++j) {
                const float sw = sigmoidf(P[j * 256 + w]);
                const float sx = sigmoidf(P[j * 256 + w + 1]);
                p_[j] = sw * sx;                       // prod of sigmoids (w, w+1)
                q_[j] = (1.f - sw) * (1.f - sx);       // prod of 1-sigmoids
                c_[j] = T[j * 256 + w] + 2.f * T[j * 256 + w + 1];  // code partial
            }
#pragma unroll
            for (int jp = 0; jp < DS; ++jp) {
                const float rf = p_[jp] * p_[jp + 1];  // 2x2 (d,w) footprint, row h
                const float rb = q_[jp] * q_[jp + 1];
                const int   a  = (int)(c_[jp] + 16.f * c_[jp + 1]);
                if (h > h0 && (d0 + jp) < (D - 1)) {   // emit output row h-1
                    const float fg   = rfp[jp] * rf;
                    const float bg   = rbp[jp] * rb;
                    const int   code = ap[jp] + 4 * a;
                    const float ga   = areaS[code];
                    const float gfg  = (code == 255) ? 1.f : 0.f;
                    const float gbg  = (code == 0) ? 1.f : 0.f;
                    const float gsf  = (ga > 0.f) ? 1.f : 0.f;
                    const float surf = 1.f - fg - bg;
                    acc[0] += fg;  acc[1] += gfg;  acc[2] += fg * gfg;
                    acc[3] += bg;  acc[4] += gbg;  acc[5] += bg * gbg;
                    acc[6] += surf * ga;
                    acc[7] += (surf + gsf) * ga;
                }
                rfp[jp] = rf; rbp[jp] = rb; ap[jp] = a;
            }
        }
    }

    // ---- Block reduction ----
    // 1) wave32 shuffle reduction of each accumulator.
#pragma unroll
    for (int i = 0; i < 8; ++i) {
        float v = acc[i];
#pragma unroll
        for (int off = 16; off > 0; off >>= 1) v += __shfl_down(v, off, 32);
        if ((t & 31) == 0) Pws[t >> 5][i] = v;
    }
    __syncthreads();

    // 2) Final 8x8 reduction as a matmul on the WMMA pipe:
    //    D[m][n] = sum_k A[m][k]*1, with A[m][k] = Pws[k + 4*pass][m]
    //    -> after two chained WMMAs, D[m][*] = total sum of accumulator m.
    if (t < 32) {
        const int m  = t & 15;
        const int hi = t >> 4;  // lanes 16..31 carry K=2/3 per the 16x4 f32 A layout
        v2f A0, A1, Bones;
        A0.x = (m < 8) ? Pws[hi * 2 + 0][m] : 0.f;
        A0.y = (m < 8) ? Pws[hi * 2 + 1][m] : 0.f;
        A1.x = (m < 8) ? Pws[4 + hi * 2 + 0][m] : 0.f;
        A1.y = (m < 8) ? Pws[4 + hi * 2 + 1][m] : 0.f;
        Bones.x = 1.0f; Bones.y = 1.0f;
        v8f Dc = {};
        Dc = __builtin_amdgcn_wmma_f32_16x16x4_f32(false, A0, false, Bones,
                                                   (short)0, Dc, false, false);
        Dc = __builtin_amdgcn_wmma_f32_16x16x4_f32(false, A1, false, Bones,
                                                   (short)0, Dc, false, false);
        if (t == 0) {
            float* wsB = ws + b * 8;
#pragma unroll
            for (int i = 0; i < 8; ++i) atomicAdd(&wsB[i], Dc[i]);
        }
    }
}

__global__ void surf_dice_final(const float* __restrict__ ws, float* __restrict__ out) {
    if (threadIdx.x == 0 && blockIdx.x == 0) {
        const float eps = 1e-5f;
        float dsum = 0.f;
#pragma unroll
        for (int b = 0; b < 2; ++b) {
            const float* s = ws + b * 8;
            const float fgd = (2.f * s[2] + eps) / (s[0] + s[1] + eps);
            const float bgd = (2.f * s[5] + eps) / (s[3] + s[4] + eps);
            const float sfd = (2.f * s[6] + eps) / (s[7] + eps);
            dsum += (fgd + bgd + sfd) * (1.f / 3.f);
        }
        out[0] = 1.f - 0.5f * dsum;
    }
}

extern "C" void kernel_launch(void* const* d_in, const int* in_sizes, int n_in,
                              void* d_out, int out_size, void* d_ws, size_t ws_size,
                              hipStream_t stream) {
    const float* preds   = (const float*)d_in[0];
    const float* targets = (const float*)d_in[1];
    // d_in[2] = power table (implicit in code construction), d_in[3] = ones kernel (implicit)
    const float* area    = (const float*)d_in[4];
    float* ws = (float*)d_ws;

    hipMemsetAsync(ws, 0, 16 * sizeof(float), stream);
    surf_dice_main<<<512, 256, 0, stream>>>(preds, targets, area, ws);
    surf_dice_final<<<1, 64, 0, stream>>>(ws, (float*)d_out);
}